// GCNLayer_64561948393627
// MI455X (gfx1250) — compile-verified
//
#include <hip/hip_runtime.h>

typedef __attribute__((ext_vector_type(2))) float v2f;
typedef __attribute__((ext_vector_type(8))) float v8f;

#define N_NODES 50000
#define N_EDGES 800000
#define FEATS   128

// ---------------------------------------------------------------------------
// Stage 0: zero the aggregation workspace (25.6 MB in d_ws), float4 stores.
// ---------------------------------------------------------------------------
__global__ void gcn_zero_kernel(float* __restrict__ agg, int n) {
    int idx = (blockIdx.x * blockDim.x + threadIdx.x) * 4;
    if (idx < n) {
        *(float4*)(agg + idx) = make_float4(0.f, 0.f, 0.f, 0.f);
    }
}

// ---------------------------------------------------------------------------
// Stage 1: edge scatter-add. One wave (32 lanes) per edge; each lane moves a
// float4 (32 x 4 = 128 feats). Non-returning f32 atomics land in L2, and the
// whole 25.6 MB accumulator is L2-resident on MI455X (192 MB L2).
// ---------------------------------------------------------------------------
__global__ void gcn_scatter_kernel(const float* __restrict__ feat,
                                   const int*   __restrict__ src,
                                   const int*   __restrict__ dst,
                                   float*       __restrict__ agg) {
    const int wave = (blockIdx.x * blockDim.x + threadIdx.x) >> 5;
    const int lane = threadIdx.x & 31;
    if (wave >= N_EDGES) return;

    const int s = src[wave];
    const int d = dst[wave];

    const float4 v = *(const float4*)(feat + (size_t)s * FEATS + lane * 4);
    float* o = agg + (size_t)d * FEATS + lane * 4;
    atomicAdd(o + 0, v.x);
    atomicAdd(o + 1, v.y);
    atomicAdd(o + 2, v.z);
    atomicAdd(o + 3, v.w);
}

// ---------------------------------------------------------------------------
// Stage 2: out = agg @ W^T + b via V_WMMA_F32_16X16X4_F32.
// One wave per 16x16 output tile. Block = 8 waves = all 8 N-tiles of one
// M-tile (shares A rows across waves via cache).
//
// A 16x4 layout : lane l (half h=l/16, lx=l%16): {A[lx][k0+2h], A[lx][k0+2h+1]}
// B 4x16 layout : {B[k0+2h][lx], B[k0+2h+1][lx]} = {W[n][k0+2h], W[n][k0+2h+1]}
//                 -> both are contiguous float2 loads from row-major storage.
// C/D layout    : VGPR r -> row m0 + r + 8h, col n0 + lx.
// ---------------------------------------------------------------------------
__global__ void gcn_gemm_wmma_kernel(const float* __restrict__ agg,
                                     const float* __restrict__ W,
                                     const float* __restrict__ bias,
                                     float*       __restrict__ out) {
    const int mt   = blockIdx.x;          // 0..3124
    const int nt   = threadIdx.x >> 5;    // 0..7
    const int lane = threadIdx.x & 31;
    const int lx   = lane & 15;
    const int hi   = lane >> 4;           // 0 or 1

    const int m0 = mt * 16;
    const int n0 = nt * 16;

    const float* aRow = agg + (size_t)(m0 + lx) * FEATS + 2 * hi;
    const float* bRow = W   + (size_t)(n0 + lx) * FEATS + 2 * hi;

    v8f c = {};
#pragma unroll
    for (int k = 0; k < FEATS; k += 4) {
        v2f a = *(const v2f*)(aRow + k);
        v2f b = *(const v2f*)(bRow + k);
        // 8 args: (neg_a, A, neg_b, B, c_mod, C, reuse_a, reuse_b)
        c = __builtin_amdgcn_wmma_f32_16x16x4_f32(
                /*neg_a=*/false, a, /*neg_b=*/false, b,
                /*c_mod=*/(short)0, c, /*reuse_a=*/false, /*reuse_b=*/false);
    }

    const float bv = bias[n0 + lx];
#pragma unroll
    for (int r = 0; r < 8; ++r) {
        out[(size_t)(m0 + r + 8 * hi) * FEATS + (n0 + lx)] = c[r] + bv;
    }
}

// ---------------------------------------------------------------------------
// Launch: inputs are (feature, src, dst, W, b) per setup_inputs() order.
// d_ws holds the [50000,128] f32 accumulator (25.6 MB).
// ---------------------------------------------------------------------------
extern "C" void kernel_launch(void* const* d_in, const int* in_sizes, int n_in,
                              void* d_out, int out_size, void* d_ws, size_t ws_size,
                              hipStream_t stream) {
    const float* feat = (const float*)d_in[0];
    const int*   src  = (const int*)d_in[1];
    const int*   dst  = (const int*)d_in[2];
    const float* W    = (const float*)d_in[3];
    const float* bias = (const float*)d_in[4];
    float*       out  = (float*)d_out;
    float*       agg  = (float*)d_ws;

    const int aggElems = N_NODES * FEATS;                  // 6.4M floats
    gcn_zero_kernel<<<(aggElems / 4 + 255) / 256, 256, 0, stream>>>(agg, aggElems);

    const int scatterThreads = N_EDGES * 32;               // one wave per edge
    gcn_scatter_kernel<<<(scatterThreads + 255) / 256, 256, 0, stream>>>(feat, src, dst, agg);

    gcn_gemm_wmma_kernel<<<N_NODES / 16, 256, 0, stream>>>(agg, W, bias, out);
}